// GCN_21672404975689
// MI455X (gfx1250) — compile-verified
//
#include <hip/hip_runtime.h>

// GCN forward, 2 layers. N=100000, E=1600000, IN=128, HID=128, OUT=64.
// fp32 end-to-end using V_WMMA_F32_16X16X4_F32 (full-precision matrix path).
// Weight matrices staged into LDS via the Tensor Data Mover (TENSOR_LOAD_TO_LDS).

typedef __attribute__((ext_vector_type(2))) float v2f;
typedef __attribute__((ext_vector_type(8))) float v8f;
typedef __attribute__((ext_vector_type(4))) unsigned int u32x4;
typedef __attribute__((ext_vector_type(8))) int i32x8;
typedef __attribute__((ext_vector_type(4))) int i32x4;

#define LANES 32

// ---------------------------------------------------------------------------
// TDM: DMA a contiguous run of `nelem` fp32 values from global to LDS offset 0.
// D# layout per CDNA5 ISA ch.8: group0 {count=1, lds_addr, global_addr, type=2},
// group1 {data_size=4B, tensor_dim0=tile_dim0=nelem, dim0_stride=nelem},
// groups 2/3 zero (<=2D tensor).
// ---------------------------------------------------------------------------
__device__ __forceinline__ void tdm_load_f32_to_lds(const void* gsrc,
                                                    unsigned lds_byte_off,
                                                    unsigned nelem) {
  unsigned long long ga = (unsigned long long)gsrc;
  u32x4 g0;
  g0[0] = 1u;                                     // count=1, user desc, no gather
  g0[1] = lds_byte_off;                           // lds_addr
  g0[2] = (unsigned)(ga & 0xFFFFFFFFu);           // global_addr[31:0]  (bits 95:64)
  g0[3] = (unsigned)((ga >> 32) & 0x01FFFFFFu)    // global_addr[56:32] (bits 120:96)
          | (2u << 30);                           // type = 2 ("image")

  i32x8 g1;
  g1[0] = (int)(2u << 16);                        // data_size=2 -> 4 bytes
  g1[1] = (int)((nelem & 0xFFFFu) << 16);         // tensor_dim0[15:0]  (bits 63:48)
  g1[2] = (int)(((nelem >> 16) & 0xFFFFu)         // tensor_dim0[31:16] (bits 79:64)
                | (1u << 16));                    // tensor_dim1 = 1    (bits 95:80)
  g1[3] = (int)((nelem & 0xFFFFu) << 16);         // tile_dim0 = nelem  (bits 127:112)
  g1[4] = 0;                                      // tile_dim1/2 unused
  g1[5] = (int)nelem;                             // tensor_dim0_stride[31:0]
  g1[6] = 0;
  g1[7] = 0;

  i32x4 z4 = {};
#if __has_include(<hip/amd_detail/amd_gfx1250_TDM.h>)
  i32x8 z8 = {};
  __builtin_amdgcn_tensor_load_to_lds(g0, g1, z4, z4, z8, 0);
#else
  __builtin_amdgcn_tensor_load_to_lds(g0, g1, z4, z4, 0);
#endif
}

// ---------------------------------------------------------------------------
// degree accumulation: deg[dst[e]] += 1  (self-loop +1 folded into rsqrt later)
// ---------------------------------------------------------------------------
__global__ void deg_kernel(const long long* __restrict__ dst, float* __restrict__ deg,
                           long long E) {
  long long t = (long long)blockIdx.x * blockDim.x + threadIdx.x;
  if (t < E) atomicAdd(&deg[dst[t]], 1.0f);
}

__global__ void dinv_kernel(float* __restrict__ deg, int N) {
  int i = blockIdx.x * blockDim.x + threadIdx.x;
  if (i < N) deg[i] = rsqrtf(deg[i] + 1.0f);  // +1 = self loop; deg+1 >= 1 always
}

// ---------------------------------------------------------------------------
// C[nrows x ncols] = A[nrows x K] @ B[K x ncols], fp32 WMMA 16x16x4.
// One wave per 16x16 output tile; blockDim.x = 32 * (ncols/16).
// B staged in LDS by the Tensor Data Mover (wave 0 issues, waits TENSORcnt,
// workgroup barrier publishes). Fragment layouts per CDNA5 ISA 7.12.2:
//   A 16x4 : lanes 0-15 hold M=0..15 {K=0,K=1}; lanes 16-31 hold {K=2,K=3}
//   B 4x16 : lanes 0-15 hold N=0..15 rows {K=0,K=1}; lanes 16-31 rows {K=2,K=3}
//   C/D    : v[i] = row (m*16 + half*8 + i), col = n*16 + (lane&15)
// ---------------------------------------------------------------------------
__global__ void gemm_wmma_kernel(const float* __restrict__ A,
                                 const float* __restrict__ B,
                                 float* __restrict__ C,
                                 int K, int ncols) {
  extern __shared__ float bs[];  // K * ncols floats, at LDS offset 0 (no static LDS)
  const int tid = threadIdx.x;

  const int lane = tid & (LANES - 1);
  const int nblk = tid >> 5;           // which 16-col block this wave owns
  const int m    = blockIdx.x;         // 16-row block
  const int half = lane >> 4;          // 0: K lo pair, 1: K hi pair
  const int l16  = lane & 15;
  const int row  = m * 16 + l16;
  const int col  = nblk * 16 + l16;
  const int koff = half * 2;

  const float* __restrict__ arow = A + (size_t)row * K;
  __builtin_prefetch(arow, 0, 0);      // global_prefetch_b8: pull X slab toward WGP

  if (tid < LANES) {                   // one TDM op per workgroup (wave 0)
    tdm_load_f32_to_lds(B, 0u, (unsigned)(K * ncols));
    __builtin_amdgcn_s_wait_tensorcnt(0);
  }
  __syncthreads();

  v8f acc = {};
  for (int k = 0; k < K; k += 4) {
    v2f a, b;
    a[0] = arow[k + koff];
    a[1] = arow[k + koff + 1];
    b[0] = bs[(k + koff) * ncols + col];
    b[1] = bs[(k + koff + 1) * ncols + col];
    acc = __builtin_amdgcn_wmma_f32_16x16x4_f32(
        /*neg_a=*/false, a, /*neg_b=*/false, b,
        /*c_mod=*/(short)0, acc, /*reuse_a=*/false, /*reuse_b=*/false);
  }

  const int row_base = m * 16 + half * 8;
  float* __restrict__ cp = C + (size_t)row_base * ncols + col;
#pragma unroll
  for (int i = 0; i < 8; ++i) cp[(size_t)i * ncols] = acc[i];
}

// ---------------------------------------------------------------------------
// Edge-parallel scatter: one wave32 per work item (E edges then N self-loops).
// out[dst] += h[src] * dinv[src]*dinv[dst], atomically. VEC = nfeat/32.
// ---------------------------------------------------------------------------
template <int VEC>
__global__ void aggregate_kernel(const float* __restrict__ h,
                                 const long long* __restrict__ src,
                                 const long long* __restrict__ dst,
                                 const float* __restrict__ dinv,
                                 float* __restrict__ out,
                                 int nfeat, long long E, int N) {
  const int lane = threadIdx.x & (LANES - 1);
  const long long wid =
      (long long)blockIdx.x * (blockDim.x >> 5) + (threadIdx.x >> 5);
  if (wid >= E + (long long)N) return;

  long long s, d;
  float w;
  if (wid < E) {
    s = src[wid];
    d = dst[wid];
    w = dinv[s] * dinv[d];
  } else {
    s = d = wid - E;
    float di = dinv[s];
    w = di * di;  // self-loop norm
  }

  const float* __restrict__ hp = h + (size_t)s * nfeat + lane * VEC;
  float* __restrict__ op = out + (size_t)d * nfeat + lane * VEC;
  if constexpr (VEC == 4) {
    float4 v = *(const float4*)hp;  // global_load_b128
    atomicAdd(op + 0, v.x * w);
    atomicAdd(op + 1, v.y * w);
    atomicAdd(op + 2, v.z * w);
    atomicAdd(op + 3, v.w * w);
  } else {
    float2 v = *(const float2*)hp;  // global_load_b64
    atomicAdd(op + 0, v.x * w);
    atomicAdd(op + 1, v.y * w);
  }
}

// ---------------------------------------------------------------------------
// h[i*nfeat+f] = (relu?) h + bias[f]
// ---------------------------------------------------------------------------
__global__ void bias_act_kernel(float* __restrict__ h, const float* __restrict__ b,
                                int nfeat, long long total, int do_relu) {
  long long t = (long long)blockIdx.x * blockDim.x + threadIdx.x;
  if (t >= total) return;
  int f = (int)(t % nfeat);
  float v = h[t] + b[f];
  h[t] = do_relu ? fmaxf(v, 0.0f) : v;
}

// ---------------------------------------------------------------------------
extern "C" void kernel_launch(void* const* d_in, const int* in_sizes, int n_in,
                              void* d_out, int out_size, void* d_ws, size_t ws_size,
                              hipStream_t stream) {
  const float*     x    = (const float*)d_in[0];
  const long long* eidx = (const long long*)d_in[1];  // int64, [2, E] flat
  const float*     W1   = (const float*)d_in[2];
  const float*     b1   = (const float*)d_in[3];
  const float*     W2   = (const float*)d_in[4];
  const float*     b2   = (const float*)d_in[5];
  float*           out  = (float*)d_out;

  constexpr int IN = 128, HID = 128, OUT = 64;
  const int       N = in_sizes[0] / IN;           // 100000
  const long long E = (long long)in_sizes[1] / 2; // 1600000
  const long long* src = eidx;
  const long long* dst = eidx + E;

  float* ws   = (float*)d_ws;
  float* dinv = ws;                          // N        (deg -> dinv in place)
  float* hlin = dinv + N;                    // N * HID  (x@W1)
  float* hagg = hlin + (size_t)N * HID;      // N * HID  (aggregated, then relu'd h)
  float* h2   = hagg + (size_t)N * HID;      // N * OUT  (h@W2)

  // ---- norm coefficients --------------------------------------------------
  hipMemsetAsync(dinv, 0, (size_t)N * sizeof(float), stream);
  {
    long long blocks = (E + 255) / 256;
    deg_kernel<<<(unsigned)blocks, 256, 0, stream>>>(dst, dinv, E);
    dinv_kernel<<<(N + 255) / 256, 256, 0, stream>>>(dinv, N);
  }

  // ---- layer 1: hlin = x @ W1 --------------------------------------------
  gemm_wmma_kernel<<<N / 16, 32 * (HID / 16), IN * HID * sizeof(float), stream>>>(
      x, W1, hlin, IN, HID);

  // ---- layer 1 aggregation + bias + relu ---------------------------------
  hipMemsetAsync(hagg, 0, (size_t)N * HID * sizeof(float), stream);
  {
    long long items = E + N;
    long long blocks = (items + 7) / 8;  // 8 waves / block of 256
    aggregate_kernel<4><<<(unsigned)blocks, 256, 0, stream>>>(
        hlin, src, dst, dinv, hagg, HID, E, N);
    long long total = (long long)N * HID;
    bias_act_kernel<<<(unsigned)((total + 255) / 256), 256, 0, stream>>>(
        hagg, b1, HID, total, /*relu=*/1);
  }

  // ---- layer 2: h2 = h @ W2 ----------------------------------------------
  gemm_wmma_kernel<<<N / 16, 32 * (OUT / 16), HID * OUT * sizeof(float), stream>>>(
      hagg, W2, h2, HID, OUT);

  // ---- layer 2 aggregation + bias ----------------------------------------
  hipMemsetAsync(out, 0, (size_t)N * OUT * sizeof(float), stream);
  {
    long long items = E + N;
    long long blocks = (items + 7) / 8;
    aggregate_kernel<2><<<(unsigned)blocks, 256, 0, stream>>>(
        h2, src, dst, dinv, out, OUT, E, N);
    long long total = (long long)N * OUT;
    bias_act_kernel<<<(unsigned)((total + 255) / 256), 256, 0, stream>>>(
        out, b2, OUT, total, /*relu=*/0);
  }
}